// SudokuUniqueHVBox_77034533421712
// MI455X (gfx1250) — compile-verified
//
#include <hip/hip_runtime.h>
#include <stdint.h>

// SudokuUniqueHVBox (type='h') for mask [B, 9, 9, 9] f32.
// Memory-bound (382 MB traffic @ 23.3 TB/s ~ 16 us floor). Strategy:
//   TDM tensor_load_to_lds (one D# per block) -> in-LDS compute ->
//   LDS --async b128--> global (ASYNCcnt path).

#define BOARD        729                     // 9*9*9 floats per board
#define TILE_BOARDS  16
#define TILE_ELEMS   (TILE_BOARDS * BOARD)   // 11664 floats = 46656 B (mult of 16)
#define BLOCK        256

typedef __attribute__((ext_vector_type(4))) unsigned int u32x4;
typedef __attribute__((ext_vector_type(8))) unsigned int u32x8;

__global__ __launch_bounds__(BLOCK) void sudoku_uniq_kernel(
    const float* __restrict__ in, float* __restrict__ out, int total)
{
    __shared__ __align__(16) float lds[TILE_ELEMS];

    const int tid = threadIdx.x;
    const long long gbase = (long long)blockIdx.x * TILE_ELEMS;
    int tileElems = total - (int)gbase;
    if (tileElems <= 0) return;                 // uniform across block
    if (tileElems > TILE_ELEMS) tileElems = TILE_ELEMS;

    const uint32_t lbase = (uint32_t)(uintptr_t)(&lds[0]);
    const int nvec = tileElems >> 2;            // 16B packets of the valid region

    // ---------------- Phase 1: TDM load  global -> LDS ----------------
    // One tensor_load_to_lds per block, issued by wave 0 (EXEC ignored by TDM).
    if (tid == 0) {
        const uint64_t gaddr = (uint64_t)(uintptr_t)(in + gbase);
        const uint32_t td0   = (uint32_t)tileElems;   // tensor_dim0: valid elems (OOB -> 0)
        const uint32_t tile0 = (uint32_t)TILE_ELEMS;  // tile_dim0 (<= 65535)

        // D# group 0 (128b): count=1 | lds_addr | global_addr[56:0] | type=2
        u32x4 g0;
        g0.x = 1u;                                               // count=1, user desc
        g0.y = lbase;                                            // lds_addr (bytes)
        g0.z = (uint32_t)gaddr;                                  // global_addr[31:0]
        g0.w = (uint32_t)((gaddr >> 32) & 0x01FFFFFFull)         // global_addr[56:32]
               | (2u << 30);                                     // type=2 ("image")
        // D# group 1 (256b): data_size=4B, 1 x tile0 tile, tensor_dim0=td0, dim1=1
        u32x8 g1;
        g1.s0 = (2u << 16);                                      // wg_mask=0, data_size=2(4B)
        g1.s1 = (td0 & 0xFFFFu) << 16;                           // abar=0 | tensor_dim0[15:0]
        g1.s2 = (td0 >> 16) | (1u << 16);                        // tensor_dim0[31:16] | tensor_dim1=1 (lo)
        g1.s3 = (tile0 << 16);                                   // tensor_dim1 hi=0 | tile_dim0
        g1.s4 = 1u;                                              // tile_dim1=1 | tile_dim2=0
        g1.s5 = tile0;                                           // tensor_dim0_stride[31:0]
        g1.s6 = 0u;                                              // stride0[47:32] | stride1 lo
        g1.s7 = 0u;                                              // stride1 hi
        asm volatile("tensor_load_to_lds %0, %1"
                     :: "s"(g0), "s"(g1) : "memory");
        asm volatile("s_wait_tensorcnt 0x0" ::: "memory");
    }
    __syncthreads();

    // ---------------- Phase 2: compute in LDS, in place ----------------
    // One thread per (board, h): owns lds[bl*729 + c*81 + h*9 + w] for all c,w.
    {
        const int nb = tileElems / BOARD;
        const int bl = tid / 9;
        const int h  = tid - bl * 9;
        if (bl < nb) {
            float* p = &lds[bl * BOARD + h * 9];
            float ud[9], acc[9];
#pragma unroll
            for (int w = 0; w < 9; ++w) acc[w] = 0.0f;
#pragma unroll
            for (int c = 0; c < 9; ++c) {
                float m[9];
                float s = 0.0f;
#pragma unroll
                for (int w = 0; w < 9; ++w) { m[w] = p[c * 81 + w]; s += m[w]; }
                // NNCompare(s,1) = min(relu(s), relu(2 - s))
                float u = fminf(fmaxf(s, 0.0f), fmaxf(2.0f - s, 0.0f));
                ud[c] = u;
#pragma unroll
                for (int w = 0; w < 9; ++w) acc[w] += fminf(u, m[w]);  // not_uniq sum over c
            }
            float cell[9];
#pragma unroll
            for (int w = 0; w < 9; ++w) cell[w] = fmaxf(1.0f - acc[w], 0.0f);  // relu(1 - sum)
#pragma unroll
            for (int c = 0; c < 9; ++c) {
#pragma unroll
                for (int w = 0; w < 9; ++w) {
                    float m  = p[c * 81 + w];
                    float mg = fmaxf(cell[w], ud[c]);          // max(cell, uniq)
                    mg = fminf(fmaxf(mg, 0.0f), 1.0f);         // clip 0..1
                    float o  = fminf(m, mg);                   // min(mask, merged)
                    p[c * 81 + w] = fminf(fmaxf(o, 0.0f), 1.0f);
                }
            }
        }
    }
    __syncthreads();

    // ---------------- Phase 3: async LDS -> global (ASYNCcnt path) ----------------
    {
        const uint64_t gdst = (uint64_t)(uintptr_t)(out + gbase);
        for (int v = tid; v < nvec; v += BLOCK) {
            uint64_t ga = gdst + (uint64_t)v * 16u;
            uint32_t la = lbase + (uint32_t)v * 16u;
            asm volatile("global_store_async_from_lds_b128 %0, %1, off"
                         :: "v"(ga), "v"(la) : "memory");
        }
        for (int e = (nvec << 2) + tid; e < tileElems; e += BLOCK) {  // tail (none for B=65536)
            uint64_t ga = gdst + (uint64_t)e * 4u;
            uint32_t la = lbase + (uint32_t)e * 4u;
            asm volatile("global_store_async_from_lds_b32 %0, %1, off"
                         :: "v"(ga), "v"(la) : "memory");
        }
    }
    asm volatile("s_wait_asynccnt 0x0" ::: "memory");
    // S_ENDPGM performs an implicit wait-idle as well.
}

extern "C" void kernel_launch(void* const* d_in, const int* in_sizes, int n_in,
                              void* d_out, int out_size, void* d_ws, size_t ws_size,
                              hipStream_t stream) {
    const float* mask = (const float*)d_in[0];
    float* out = (float*)d_out;
    const int total = in_sizes[0];                       // 65536 * 729
    const int blocks = (total + TILE_ELEMS - 1) / TILE_ELEMS;
    sudoku_uniq_kernel<<<blocks, BLOCK, 0, stream>>>(mask, out, total);
}